// QWENWrapper_1778116461255
// MI455X (gfx1250) — compile-verified
//
#include <hip/hip_runtime.h>
#include <hip/hip_bf16.h>
#include <math.h>

// ---------------- model constants (match reference) ----------------
#define Lc      2
#define NH      16
#define NKV     8
#define HD      128
#define Hc      1024
#define Ic      3072
#define Vc      32000
#define HIST    2048
#define IDSc    1
#define KV_SEQ  (HIST + IDSc)          // 2049
#define EPSc    1e-6f

// ---------------- workspace layout (float offsets) -----------------
#define WS_HS      0                    // 1024
#define WS_HN      1024                 // 1024
#define WS_Q       2048                 // 2048
#define WS_K       4096                 // 1024
#define WS_V       5120                 // 1024
#define WS_QR      6144                 // 2048
#define WS_SCORES  8192                 // 16*2049 = 32784
#define WS_AO      40976                // 2048
#define WS_G       43024                // 3072
#define WS_U       46096                // 3072
#define WS_LOGITS  49168                // 32000
#define WS_PVAL    81168                // 125
#define WS_PIDX    81296                // 125 (as int)

typedef __attribute__((ext_vector_type(2))) float v2f;
typedef __attribute__((ext_vector_type(8))) float v8f;
typedef __attribute__((ext_vector_type(4))) unsigned int v4u;
typedef __attribute__((ext_vector_type(4))) int v4i;
typedef __attribute__((ext_vector_type(8))) int v8i;

#if __has_builtin(__builtin_amdgcn_tensor_load_to_lds) && \
    __has_builtin(__builtin_amdgcn_s_wait_tensorcnt)
#define USE_TDM 1
#endif

// ----------------------------------------------------------------------
// TDM stage: DMA `xlen` fp32 elements from global `x` into LDS `xs`
// using TENSOR_LOAD_TO_LDS (ISA ch.7/8: D# group0 + group1, 2-D tensor).
// Must be called wave-uniformly; caller barriers afterwards.
// ----------------------------------------------------------------------
#ifdef USE_TDM
__device__ __forceinline__
void tdm_load_vec_to_lds(const float* x, float* xs_generic, int xlen)
{
    unsigned long long ga = (unsigned long long)(size_t)x;       // global VA
    unsigned lds_off = (unsigned)(size_t)(void*)xs_generic;      // addr[31:0] = LDS offset

    v4u g0;
    g0[0] = 1u;                                                   // count=1 (valid), user mode
    g0[1] = lds_off;                                              // lds_addr
    g0[2] = (unsigned)(ga & 0xffffffffu);                         // global_addr[31:0]
    g0[3] = (unsigned)((ga >> 32) & 0x01ffffffu) | 0x80000000u;   // global_addr[56:32] | type=2

    v8i g1;
    g1[0] = 0x00020000;                        // workgroup_mask=0, data_size=2 (4B)
    g1[1] = (xlen & 0xffff) << 16;             // tensor_dim0[15:0]   (bits 63:48)
    g1[2] = ((xlen >> 16) & 0xffff) | (1 << 16); // tensor_dim0[31:16], tensor_dim1 lo = 1
    g1[3] = (xlen & 0xffff) << 16;             // tensor_dim1 hi = 0, tile_dim0 = xlen
    g1[4] = 1;                                 // tile_dim1 = 1, tile_dim2 = 0
    g1[5] = xlen;                              // tensor_dim0_stride[31:0]
    g1[6] = 0;                                 // stride hi, tensor_dim1_stride lo
    g1[7] = 0;

    v4i gz = {0, 0, 0, 0};
#if defined(__clang_major__) && (__clang_major__ >= 23)
    v8i gz8 = {0, 0, 0, 0, 0, 0, 0, 0};
    __builtin_amdgcn_tensor_load_to_lds(g0, g1, gz, gz, gz8, 0);
#else
    __builtin_amdgcn_tensor_load_to_lds(g0, g1, gz, gz, 0);
#endif
    __builtin_amdgcn_s_wait_tensorcnt(0);
}
#endif

// ======================================================================
// GEMV via V_WMMA_F32_16X16X4_F32.
// out[m] = sum_h W[m][h] * x[h]   for M outputs, W row-major (M x xlen).
// One wave -> 16 outputs (N dim of B). A holds x broadcast over all 16 M
// rows (row 0 is the one we read back). Per issue the wave consumes a
// 4x16 fp32 weight tile (256B): pure streaming of the weight matrix.
// x vector is staged into LDS by the Tensor Data Mover (wave 0 issues,
// s_wait_tensorcnt 0, then block barrier).
// mode 0: out[m]=r ; mode 1: out[m]+=r (residual add)
// ======================================================================
__global__ __launch_bounds__(128)
void gemv_wmma_f32(const float* __restrict__ W, const float* __restrict__ x,
                   float* __restrict__ out, int xlen, int M, int mode)
{
    __shared__ float xs[Ic];            // up to 3072 floats = 12KB
    const int tid = threadIdx.x;

#ifdef USE_TDM
    if ((tid >> 5) == 0) {              // wave-uniform: only wave 0 issues the DMA
        tdm_load_vec_to_lds(x, xs, xlen);
    }
    __syncthreads();
#else
    for (int j = tid; j < xlen; j += 128) xs[j] = x[j];
    __syncthreads();
#endif

    const int wave = tid >> 5;
    const int lane = tid & 31;
    const int m0   = (blockIdx.x * 4 + wave) * 16;
    const int n    = lane & 15;
    const int kb   = (lane >> 4) << 1;   // 0 (lanes 0-15) or 2 (lanes 16-31)

    const float* wrow = W + (size_t)(m0 + n) * (size_t)xlen + kb;

    v8f c = {};
    #pragma unroll 4
    for (int h = 0; h < xlen; h += 4) {
        if ((h & 127) == 0)
            __builtin_prefetch(wrow + h + 512, 0, 0);   // global_prefetch_b8
        v2f b = *(const v2f*)(wrow + h);                // 8B/lane, 256B/wave/issue
        v2f a;
        a[0] = xs[h + kb];
        a[1] = xs[h + kb + 1];
        // D = A x B + C   (f32, 16x16x4)
        c = __builtin_amdgcn_wmma_f32_16x16x4_f32(false, a, false, b,
                                                  (short)0, c, false, false);
    }
    if (lane < 16) {
        float r = c[0];                  // (M=0, N=lane)
        int m = m0 + n;
        if (m < M) {
            if (mode == 0) out[m] = r;
            else           out[m] += r;
        }
    }
}

// ======================= embedding dequant gather =====================
__global__ __launch_bounds__(256)
void embed_k(const int* __restrict__ ids, const unsigned char* __restrict__ data,
             const float* __restrict__ scale, const float* __restrict__ zp,
             float* __restrict__ hs)
{
    int j = blockIdx.x * 256 + threadIdx.x;
    if (j < Hc) {
        int t = ids[0];
        hs[j] = (float)data[(size_t)t * Hc + j] * scale[t] + zp[t];
    }
}

// ============================ RMS norm ================================
__global__ __launch_bounds__(256)
void rmsnorm_k(const float* __restrict__ x, const float* __restrict__ w,
               float* __restrict__ out, int n)
{
    __shared__ float red[256];
    int tid = threadIdx.x;
    float s = 0.f;
    for (int j = tid; j < n; j += 256) { float v = x[j]; s += v * v; }
    red[tid] = s; __syncthreads();
    for (int off = 128; off > 0; off >>= 1) {
        if (tid < off) red[tid] += red[tid + off];
        __syncthreads();
    }
    float sc = rsqrtf(red[0] / (float)n + EPSc);
    for (int j = tid; j < n; j += 256) out[j] = w[j] * x[j] * sc;
}

// ========== per-head q/k RMS + RoPE, append KV column 2048 ============
__global__ __launch_bounds__(128)
void qk_postproc(const float* __restrict__ q, const float* __restrict__ k,
                 const float* __restrict__ v,
                 const float* __restrict__ q_norm, const float* __restrict__ k_norm,
                 const float* __restrict__ cos_e, const float* __restrict__ sin_e,
                 float* __restrict__ qr,
                 float* __restrict__ out_keys, float* __restrict__ out_vals,
                 int layer)
{
    __shared__ float red[128];
    __shared__ float tmp[128];
    const int d = threadIdx.x;
    const float cv = cos_e[(size_t)HIST * HD + d];   // position = history_len
    const float sv = sin_e[(size_t)HIST * HD + d];

    // ---- Q heads ----
    for (int h = 0; h < NH; h++) {
        float x = q[h * HD + d];
        red[d] = x * x; __syncthreads();
        for (int off = 64; off > 0; off >>= 1) {
            if (d < off) red[d] += red[d + off];
            __syncthreads();
        }
        float sc = rsqrtf(red[0] / (float)HD + EPSc);
        __syncthreads();
        float qn = x * sc * q_norm[d];
        tmp[d] = qn; __syncthreads();
        float rot = (d < 64) ? -tmp[d + 64] : tmp[d - 64];
        qr[h * HD + d] = qn * cv + rot * sv;
        __syncthreads();
    }
    // ---- KV heads: rope key, append key/value at position KV_SEQ-1 ----
    for (int kh = 0; kh < NKV; kh++) {
        float x = k[kh * HD + d];
        red[d] = x * x; __syncthreads();
        for (int off = 64; off > 0; off >>= 1) {
            if (d < off) red[d] += red[d + off];
            __syncthreads();
        }
        float sc = rsqrtf(red[0] / (float)HD + EPSc);
        __syncthreads();
        float kn = x * sc * k_norm[d];
        tmp[d] = kn; __syncthreads();
        float rot = (d < 64) ? -tmp[d + 64] : tmp[d - 64];
        float kr = kn * cv + rot * sv;
        size_t kbase = ((size_t)(layer * NKV + kh) * HD + d) * KV_SEQ;
        out_keys[kbase + (KV_SEQ - 1)] = kr;
        size_t vbase = ((size_t)(layer * NKV + kh) * KV_SEQ + (KV_SEQ - 1)) * HD + d;
        out_vals[vbase] = v[kh * HD + d];
        __syncthreads();
    }
}

// ====== copy past KV into output (stride 2048 -> 2049 re-layout) ======
__global__ __launch_bounds__(256)
void copy_past(const float* __restrict__ pk, const float* __restrict__ pv,
               float* __restrict__ ok, float* __restrict__ ov)
{
    size_t idx = (size_t)blockIdx.x * 256 + threadIdx.x;   // over L*NKV*HD*HIST
    // keys: [l][kh][d][p]  -> dst row stride KV_SEQ
    {
        size_t p = idx % HIST, r = idx / HIST;
        ok[r * KV_SEQ + p] = pk[idx];
    }
    // values: [l][kh][p][d] -> dst pos stride KV_SEQ
    {
        size_t d = idx % HD, r = idx / HD;
        size_t p = r % HIST, b = r / HIST;
        ov[(b * KV_SEQ + p) * HD + d] = pv[idx];
    }
}

// =================== attention: scores = q.K + mask ===================
__global__ __launch_bounds__(256)
void attn_scores(const float* __restrict__ qr, const float* __restrict__ keys,
                 const float* __restrict__ attn_flag, float* __restrict__ scores)
{
    __shared__ float qs[HD];
    const int h = blockIdx.y;
    const int p = blockIdx.x * 256 + threadIdx.x;
    if (threadIdx.x < HD) qs[threadIdx.x] = qr[h * HD + threadIdx.x];
    __syncthreads();
    if (p >= KV_SEQ) return;
    const float* krow = keys + (size_t)(h >> 1) * HD * KV_SEQ + p;  // G=2
    float s = 0.f;
    #pragma unroll 8
    for (int d = 0; d < HD; d++) s += qs[d] * krow[(size_t)d * KV_SEQ];
    // causal row 0 of mask: 0 at p==0, -128 elsewhere, scaled by attn_flag
    s += (p > 0) ? (-128.f * attn_flag[0]) : 0.f;
    scores[h * KV_SEQ + p] = s;
}

// =========================== softmax per head =========================
__global__ __launch_bounds__(256)
void softmax_k(float* __restrict__ scores)
{
    __shared__ float red[256];
    float* s = scores + blockIdx.x * KV_SEQ;
    const int tid = threadIdx.x;

    float m = -1e30f;
    for (int p = tid; p < KV_SEQ; p += 256) m = fmaxf(m, s[p]);
    red[tid] = m; __syncthreads();
    for (int off = 128; off > 0; off >>= 1) {
        if (tid < off) red[tid] = fmaxf(red[tid], red[tid + off]);
        __syncthreads();
    }
    m = red[0]; __syncthreads();

    float sum = 0.f;
    for (int p = tid; p < KV_SEQ; p += 256) {
        float e = __expf(s[p] - m); s[p] = e; sum += e;
    }
    red[tid] = sum; __syncthreads();
    for (int off = 128; off > 0; off >>= 1) {
        if (tid < off) red[tid] += red[tid + off];
        __syncthreads();
    }
    float inv = 1.f / red[0];
    for (int p = tid; p < KV_SEQ; p += 256) s[p] *= inv;
}

// ========================= out = attn @ V =============================
__global__ __launch_bounds__(128)
void attn_v(const float* __restrict__ attn, const float* __restrict__ vals,
            float* __restrict__ attn_out)
{
    const int h = blockIdx.x, d = threadIdx.x;
    const float* a    = attn + h * KV_SEQ;
    const float* vrow = vals + (size_t)(h >> 1) * KV_SEQ * HD + d;
    float acc = 0.f;
    for (int p = 0; p < KV_SEQ; p++) acc += a[p] * vrow[(size_t)p * HD];
    attn_out[h * HD + d] = acc;
}

// ======================= g = silu(g) * u ==============================
__global__ __launch_bounds__(256)
void silu_mul(float* __restrict__ g, const float* __restrict__ u, int n)
{
    int i = blockIdx.x * 256 + threadIdx.x;
    if (i < n) {
        float x = g[i];
        g[i] = (x / (1.f + __expf(-x))) * u[i];
    }
}

// ============================ argmax ==================================
__global__ __launch_bounds__(256)
void argmax_part(const float* __restrict__ logits, float* __restrict__ pv,
                 int* __restrict__ pi)
{
    __shared__ float sv[256];
    __shared__ int   si[256];
    int tid = threadIdx.x;
    int idx = blockIdx.x * 256 + tid;       // 125*256 == 32000 exactly
    sv[tid] = logits[idx]; si[tid] = idx; __syncthreads();
    for (int off = 128; off > 0; off >>= 1) {
        if (tid < off) {
            float v2 = sv[tid + off]; int i2 = si[tid + off];
            if (v2 > sv[tid] || (v2 == sv[tid] && i2 < si[tid])) { sv[tid] = v2; si[tid] = i2; }
        }
        __syncthreads();
    }
    if (tid == 0) { pv[blockIdx.x] = sv[0]; pi[blockIdx.x] = si[0]; }
}

__global__ __launch_bounds__(256)
void argmax_final(const float* __restrict__ pv, const int* __restrict__ pi,
                  int* __restrict__ tail, int nparts)
{
    __shared__ float sv[256];
    __shared__ int   si[256];
    int tid = threadIdx.x;
    sv[tid] = (tid < nparts) ? pv[tid] : -1e30f;
    si[tid] = (tid < nparts) ? pi[tid] : 0x7fffffff;
    __syncthreads();
    for (int off = 128; off > 0; off >>= 1) {
        if (tid < off) {
            float v2 = sv[tid + off]; int i2 = si[tid + off];
            if (v2 > sv[tid] || (v2 == sv[tid] && i2 < si[tid])) { sv[tid] = v2; si[tid] = i2; }
        }
        __syncthreads();
    }
    if (tid == 0) { tail[0] = si[0]; tail[1] = KV_SEQ; }
}

// ======================================================================
extern "C" void kernel_launch(void* const* d_in, const int* in_sizes, int n_in,
                              void* d_out, int out_size, void* d_ws, size_t ws_size,
                              hipStream_t stream)
{
    // ---- inputs, setup_inputs() insertion order, params expanded in place
    const float* past_keys   = (const float*)d_in[0];
    const float* past_values = (const float*)d_in[1];
    const int*   input_ids   = (const int*)  d_in[2];
    const float* attn_flag   = (const float*)d_in[3];
    //            history_len d_in[4], ids_len d_in[5] : compile-time constants
    const float* in_ln      = (const float*)d_in[6];
    const float* q_w        = (const float*)d_in[7];
    const float* k_w        = (const float*)d_in[8];
    const float* v_w        = (const float*)d_in[9];
    const float* q_norm     = (const float*)d_in[10];
    const float* k_norm     = (const float*)d_in[11];
    const float* o_w        = (const float*)d_in[12];
    const float* post_ln    = (const float*)d_in[13];
    const float* gate_w     = (const float*)d_in[14];
    const float* up_w       = (const float*)d_in[15];
    const float* down_w     = (const float*)d_in[16];
    const float* final_norm = (const float*)d_in[17];
    const float* lm_head    = (const float*)d_in[18];
    const unsigned char* embed_data = (const unsigned char*)d_in[19];
    const float* embed_scale = (const float*)d_in[20];
    const float* embed_zp    = (const float*)d_in[21];
    const float* cos_emb     = (const float*)d_in[22];
    const float* sin_emb     = (const float*)d_in[23];
    // causal_mask d_in[24]: folded into the (p>0 ? -128 : 0)*attn_flag formula

    float* ws = (float*)d_ws;
    float* hs = ws + WS_HS;    float* hn = ws + WS_HN;
    float* q  = ws + WS_Q;     float* k  = ws + WS_K;     float* v = ws + WS_V;
    float* qr = ws + WS_QR;    float* sc = ws + WS_SCORES;
    float* ao = ws + WS_AO;    float* g  = ws + WS_G;     float* u = ws + WS_U;
    float* lg = ws + WS_LOGITS;
    float* pv = ws + WS_PVAL;  int* pi = (int*)(ws + WS_PIDX);

    float* out_keys = (float*)d_out;
    float* out_vals = out_keys + (size_t)Lc * NKV * HD * KV_SEQ;   // 4,196,352
    int*   tail     = (int*)d_out + 2 * (size_t)Lc * NKV * HD * KV_SEQ;

    // KV cache re-layout copy (independent of compute; both layers)
    copy_past<<<(Lc * NKV * HD * HIST) / 256, 256, 0, stream>>>(
        past_keys, past_values, out_keys, out_vals);

    // embedding
    embed_k<<<(Hc + 255) / 256, 256, 0, stream>>>(input_ids, embed_data,
                                                  embed_scale, embed_zp, hs);

    for (int i = 0; i < Lc; i++) {
        const size_t wq = (size_t)i * (NH * HD) * Hc;
        const size_t wk = (size_t)i * (NKV * HD) * Hc;
        const size_t wo = (size_t)i * Hc * (NH * HD);
        const size_t wg = (size_t)i * Ic * Hc;
        const size_t wd = (size_t)i * Hc * Ic;

        rmsnorm_k<<<1, 256, 0, stream>>>(hs, in_ln + i * Hc, hn, Hc);

        gemv_wmma_f32<<<(NH * HD) / 64, 128, 0, stream>>>(q_w + wq, hn, q, Hc, NH * HD, 0);
        gemv_wmma_f32<<<(NKV * HD) / 64, 128, 0, stream>>>(k_w + wk, hn, k, Hc, NKV * HD, 0);
        gemv_wmma_f32<<<(NKV * HD) / 64, 128, 0, stream>>>(v_w + wk, hn, v, Hc, NKV * HD, 0);

        qk_postproc<<<1, 128, 0, stream>>>(q, k, v, q_norm + i * HD, k_norm + i * HD,
                                           cos_emb, sin_emb, qr, out_keys, out_vals, i);

        const float* lkeys = out_keys + (size_t)i * NKV * HD * KV_SEQ;
        const float* lvals = out_vals + (size_t)i * NKV * KV_SEQ * HD;

        attn_scores<<<dim3((KV_SEQ + 255) / 256, NH), 256, 0, stream>>>(qr, lkeys, attn_flag, sc);
        softmax_k<<<NH, 256, 0, stream>>>(sc);
        attn_v<<<NH, 128, 0, stream>>>(sc, lvals, ao);

        gemv_wmma_f32<<<Hc / 64, 128, 0, stream>>>(o_w + wo, ao, hs, NH * HD, Hc, 1); // residual add

        rmsnorm_k<<<1, 256, 0, stream>>>(hs, post_ln + i * Hc, hn, Hc);
        gemv_wmma_f32<<<Ic / 64, 128, 0, stream>>>(gate_w + wg, hn, g, Hc, Ic, 0);
        gemv_wmma_f32<<<Ic / 64, 128, 0, stream>>>(up_w   + wg, hn, u, Hc, Ic, 0);
        silu_mul<<<Ic / 256, 256, 0, stream>>>(g, u, Ic);
        gemv_wmma_f32<<<Hc / 64, 128, 0, stream>>>(down_w + wd, g, hs, Ic, Hc, 1);    // residual add
    }

    rmsnorm_k<<<1, 256, 0, stream>>>(hs, final_norm, hn, Hc);
    gemv_wmma_f32<<<Vc / 64, 128, 0, stream>>>(lm_head, hn, lg, Hc, Vc, 0);

    argmax_part<<<Vc / 256, 256, 0, stream>>>(lg, pv, pi);
    argmax_final<<<1, 256, 0, stream>>>(pv, pi, tail, Vc / 256);
}